// GIN_ModelBen1_45792941310039
// MI455X (gfx1250) — compile-verified
//
#include <hip/hip_runtime.h>

typedef __attribute__((ext_vector_type(2))) float v2f;
typedef __attribute__((ext_vector_type(8))) float v8f;

#define N_NODES 100000
#define N_EDGES 640000
#define D 128

// ---------------------------------------------------------------------------
// Kernel 1: edge scatter-add. One wave (32 lanes) per edge; each lane does 4
// f32 atomics at stride 32 so every atomic batch from a wave is a contiguous
// 128B segment of agg[dst] (L2-friendly, coalesced).
// ---------------------------------------------------------------------------
__global__ void __launch_bounds__(256) gin_scatter(
    const float* __restrict__ x,
    const long long* __restrict__ ei,   // [2, N_EDGES] int64: row0=src, row1=dst
    float* __restrict__ agg) {
  int gid  = blockIdx.x * 256 + threadIdx.x;
  int edge = gid >> 5;
  int lane = gid & 31;
  if (edge >= N_EDGES) return;
  long long s = ei[edge];
  long long d = ei[N_EDGES + edge];
  const float* xs = x + s * (long long)D;
  float* ad = agg + d * (long long)D;
#pragma unroll
  for (int j = 0; j < 4; ++j) {
    int f = lane + j * 32;
    atomicAdd(&ad[f], xs[f]);
  }
}

// ---------------------------------------------------------------------------
// Kernel 2: out = (x + agg) @ W^T + b using V_WMMA_F32_16X16X4_F32.
// Block = 256 threads = 8 waves; each wave computes a 16x128 output tile.
// W^T staged in LDS pair-interleaved: lds[(k>>1)*256 + 2n + (k&1)] = W[n][k],
// so each lane's B fragment (B[k0][n], B[k0+1][n]) is a single ds_load_b64.
// ---------------------------------------------------------------------------
__global__ void __launch_bounds__(256) gin_gemm(
    const float* __restrict__ x,
    const float* __restrict__ agg,
    const float* __restrict__ W,     // [D_OUT, D_IN] row-major
    const float* __restrict__ bias,  // [D_OUT]
    float* __restrict__ out) {
  __shared__ float lds[D * D];  // 64 KB (WGP has 320 KB)

  int tid = threadIdx.x;
  // Cooperative load of W, coalesced in global (k contiguous per thread group).
  for (int i = tid; i < D * D; i += 256) {
    int n = i >> 7;    // output-channel (row of W)
    int k = i & 127;   // input-channel
    lds[(k >> 1) * 256 + 2 * n + (k & 1)] = W[n * D + k];
  }
  __syncthreads();

  int lane = tid & 31;
  int wave = tid >> 5;
  int m    = lane & 15;   // row within 16-tile (A), col within 16-tile (B/D)
  int half = lane >> 4;   // K-pair select for A/B, row+8 select for D

  long long rowBase = (long long)blockIdx.x * 128 + wave * 16;
  long long arow    = rowBase + m;
  if (arow > N_NODES - 1) arow = N_NODES - 1;  // clamp reads; EXEC stays all-1s
  const float* xr = x   + arow * D;
  const float* ar = agg + arow * D;

  v8f acc[8];
#pragma unroll
  for (int t = 0; t < 8; ++t) acc[t] = 0.0f;

#pragma unroll
  for (int kk = 0; kk < D; kk += 4) {
    int k0 = kk + half * 2;
    // A fragment: h = x + agg, fused here (saves a full copy pass over HBM)
    v2f a = *(const v2f*)(xr + k0) + *(const v2f*)(ar + k0);
    const float* lrow = &lds[(k0 >> 1) * 256];
#pragma unroll
    for (int nt = 0; nt < 8; ++nt) {
      v2f b = *(const v2f*)(lrow + 2 * (nt * 16 + m));
      // 8 args: (neg_a, A, neg_b, B, c_mod, C, reuse_a, reuse_b)
      acc[nt] = __builtin_amdgcn_wmma_f32_16x16x4_f32(
          false, a, false, b, (short)0, acc[nt], false, false);
    }
  }

  // D layout: VGPR v -> row (v + half*8), lanes' col = nt*16 + m.
#pragma unroll
  for (int nt = 0; nt < 8; ++nt) {
    int col = nt * 16 + m;
    float bv = bias[col];
#pragma unroll
    for (int v = 0; v < 8; ++v) {
      long long row = rowBase + v + half * 8;
      if (row < N_NODES) out[row * D + col] = acc[nt][v] + bv;
    }
  }
}

// ---------------------------------------------------------------------------
extern "C" void kernel_launch(void* const* d_in, const int* in_sizes, int n_in,
                              void* d_out, int out_size, void* d_ws, size_t ws_size,
                              hipStream_t stream) {
  const float*     x  = (const float*)d_in[0];
  const long long* ei = (const long long*)d_in[1];  // int64 edge_index
  const float*     W  = (const float*)d_in[2];
  const float*     b  = (const float*)d_in[3];
  float* out = (float*)d_out;
  float* agg = (float*)d_ws;  // 100000*128*4 = 51.2 MB scratch

  // Zero the aggregation buffer every call (graph-capture safe).
  hipMemsetAsync(agg, 0, (size_t)N_NODES * D * sizeof(float), stream);

  // Scatter: 640000 edges * 32 lanes = 20.48M threads.
  int scatterBlocks = (N_EDGES * 32) / 256;  // 80000
  gin_scatter<<<scatterBlocks, 256, 0, stream>>>(x, ei, agg);

  // GEMM: 128 node-rows per block.
  int gemmBlocks = (N_NODES + 127) / 128;    // 782
  gin_gemm<<<gemmBlocks, 256, 0, stream>>>(x, agg, W, b, out);
}